// PentachoronCollectiveConsensus_46523085750353
// MI455X (gfx1250) — compile-verified
//
#include <hip/hip_runtime.h>
#include <math.h>

typedef __attribute__((ext_vector_type(2))) float v2f;
typedef __attribute__((ext_vector_type(8))) float v8f;

#define NVERT 5
#define DIMN  512

__device__ __forceinline__ float sigmoidf_(float x) {
    return 1.0f / (1.0f + __expf(-x));
}

// One wave computes the 5x5 Gram matrices of 3 pentachora via
// V_WMMA_F32_16X16X4_F32 (rows 0-4 / 5-9 / 10-14 of the 16-row A tile; row 15
// is a don't-care pad row whose contribution is confined to D row/col 15),
// then 3 lanes run the tiny Cayley-Menger / stats / Cantor epilogue.
__global__ __launch_bounds__(256)
void pentachoron_consensus_kernel(const float* __restrict__ P,
                                  float* __restrict__ out, int V)
{
    __shared__ float lg[8 * 256];                 // 16x16 f32 tile per wave

    const int wave      = threadIdx.x >> 5;
    const int lane      = threadIdx.x & 31;
    const int triple    = blockIdx.x * 8 + wave;  // 3 items per wave
    const int base_item = triple * 3;

    // A-matrix 16x4 f32 layout: lane holds row (lane&15); lanes 0-15 supply
    // K={0,1}, lanes 16-31 supply K={2,3} of each K-step.
    const int rr   = lane & 15;
    const int kh   = lane >> 4;
    const int sub  = rr / 5;                      // item within triple (3 => pad row)
    const int vert = rr - 5 * sub;
    const int item = base_item + sub;

    // Unconditional loads: clamp out-of-range items (incl. the rr==15 pad row)
    // to a valid address. Their Gram contributions land only in rows/cols we
    // never read, and the data is real & finite, so no zeroing is needed.
    const int safe_item = (item < V) ? item : (V - 1);
    const float* src = P + ((size_t)safe_item * NVERT + vert) * DIMN + 2 * kh;

    v8f acc = {0.f, 0.f, 0.f, 0.f, 0.f, 0.f, 0.f, 0.f};

    #pragma unroll 8
    for (int k = 0; k < DIMN; k += 4) {
        // Gram: D = A * A^T + C. For the symmetric product the per-lane B
        // registers equal the A registers (B(k,n) = A(n,k) under the 4x16
        // B striping), so pass the same v2f for both operands.
        const v2f a = *(const v2f*)(src + k);
        acc = __builtin_amdgcn_wmma_f32_16x16x4_f32(false, a, false, a,
                                                    (short)0, acc, false, false);
    }

    // C/D layout: VGPR r holds M=r (lanes 0-15) / M=r+8 (lanes 16-31), N=lane&15.
    {
        const int base = wave * 256 + (kh * 8) * 16 + rr;
        #pragma unroll
        for (int r = 0; r < 8; ++r) lg[base + r * 16] = acc[r];
    }
    __syncthreads();   // uniform across the block (identical trip counts)

    if (lane < 3) {
        const int it = base_item + lane;
        if (it < V) {
            const float* G = &lg[wave * 256 + (5 * lane) * 16 + (5 * lane)];
            float g[5][5];
            #pragma unroll
            for (int i = 0; i < 5; ++i)
                #pragma unroll
                for (int j = 0; j < 5; ++j)
                    g[i][j] = G[i * 16 + j];

            // Pairwise squared distances.
            float d2[5][5];
            #pragma unroll
            for (int i = 0; i < 5; ++i) {
                d2[i][i] = 0.f;
                #pragma unroll
                for (int j = 0; j < 5; ++j)
                    if (j != i)
                        d2[i][j] = fmaxf(g[i][i] + g[j][j] - 2.f * g[i][j], 0.f);
            }

            // Edge statistics over the 10 unique edges (unbiased std).
            float e[10];
            float esum = 0.f;
            int idx = 0;
            #pragma unroll
            for (int i = 0; i < 5; ++i)
                #pragma unroll
                for (int j = i + 1; j < 5; ++j) {
                    e[idx] = sqrtf(d2[i][j]);
                    esum += e[idx];
                    ++idx;
                }
            const float mean_e = esum * 0.1f;
            float evar = 0.f;
            #pragma unroll
            for (int q = 0; q < 10; ++q) {
                const float d = e[q] - mean_e;
                evar += d * d;
            }
            const float std_e = sqrtf(evar * (1.0f / 9.0f));

            // Cayley-Menger 6x6 determinant. Swap rows 0 and 1 up front so
            // the (0,0) pivot is 1; fully unrolled no-pivot elimination.
            float m[6][6];
            m[0][0] = 1.f;
            #pragma unroll
            for (int j = 1; j < 6; ++j) m[0][j] = d2[0][j - 1];
            m[1][0] = 0.f;
            #pragma unroll
            for (int j = 1; j < 6; ++j) m[1][j] = 1.f;
            #pragma unroll
            for (int r = 2; r < 6; ++r) {
                m[r][0] = 1.f;
                #pragma unroll
                for (int j = 1; j < 6; ++j) m[r][j] = d2[r - 1][j - 1];
            }
            float det = -1.f;                         // row-swap sign
            #pragma unroll
            for (int c = 0; c < 6; ++c) {
                const float piv = m[c][c];
                det *= piv;
                const float inv = (piv != 0.f) ? (1.f / piv) : 0.f;
                #pragma unroll
                for (int r = c + 1; r < 6; ++r) {
                    const float f = m[r][c] * inv;
                    #pragma unroll
                    for (int cc = c + 1; cc < 6; ++cc)
                        m[r][cc] -= f * m[c][cc];
                }
            }
            const float volume = sqrtf(fmaxf(-det * (1.0f / 9216.0f), 0.0f));

            // Vertex spread from centroid, entirely from the Gram matrix:
            // ||p_i - c||^2 = g_ii - 0.4*rowsum_i + 0.04*sum_all.
            float sum_all = 0.f;
            float rowsum[5];
            #pragma unroll
            for (int i = 0; i < 5; ++i) {
                float rs = 0.f;
                #pragma unroll
                for (int j = 0; j < 5; ++j) rs += g[i][j];
                rowsum[i] = rs;
                sum_all += rs;
            }
            float cd[5];
            float csum = 0.f;
            #pragma unroll
            for (int i = 0; i < 5; ++i) {
                const float c2 = g[i][i] - 0.4f * rowsum[i] + 0.04f * sum_all;
                cd[i] = sqrtf(fmaxf(c2, 0.f));
                csum += cd[i];
            }
            const float cmean = csum * 0.2f;
            float cvar = 0.f;
            #pragma unroll
            for (int i = 0; i < 5; ++i) {
                const float d = cd[i] - cmean;
                cvar += d * d;
            }
            const float spread = sqrtf(cvar * 0.25f);

            const float vn = sigmoidf_(volume * 10.f);
            const float er = sigmoidf_(std_e / (mean_e + 1e-6f));
            const float sn = sigmoidf_(spread);

            float seed  = fminf(fmaxf(0.4f * vn + 0.3f * er + 0.3f * sn,
                                      1e-6f), 1.f - 1e-6f);
            const float drift = (vn + er + sn) * 0.01f;

            float x = seed, cv = 0.f, factor = 0.5f;
            #pragma unroll
            for (int s = 0; s < 8; ++s) {
                const float xs = x * 3.f;
                const float dg = floorf(xs);
                const float xf = xs - dg;
                cv += (dg == 2.f) ? factor : 0.f;
                x = fminf(fmaxf(xf + drift, 1e-6f), 1.f - 1e-6f);
                factor *= 0.5f;
            }
            out[it] = fminf(fmaxf(cv, 0.f), 1.f);
        }
    }
}

extern "C" void kernel_launch(void* const* d_in, const int* in_sizes, int n_in,
                              void* d_out, int out_size, void* d_ws, size_t ws_size,
                              hipStream_t stream) {
    const float* P = (const float*)d_in[0];
    float* out = (float*)d_out;
    const int V = in_sizes[0] / (NVERT * DIMN);
    const int triples = (V + 2) / 3;          // 3 items per wave
    const int blocks  = (triples + 7) / 8;    // 8 waves per 256-thread block
    hipLaunchKernelGGL(pentachoron_consensus_kernel, dim3(blocks), dim3(256),
                       0, stream, P, out, V);
}